// GNNModel_71193377899389
// MI455X (gfx1250) — compile-verified
//
#include <hip/hip_runtime.h>
#include <stdint.h>

typedef float v2f __attribute__((ext_vector_type(2)));
typedef float v8f __attribute__((ext_vector_type(8)));

// ---------------------------------------------------------------------------
// Y[M,32] = X[M,K] @ W[K,32] + bias      (exact fp32 via V_WMMA_F32_16X16X4_F32)
// One wave -> 16 rows x 32 cols (two 16x16 f32 accumulators, shared A frag).
//
// A (16x4 f32) layout per ISA: lanes 0-15 hold row M=lane, VGPR0=K, VGPR1=K+1;
// lanes 16-31 hold row M=lane-16, VGPR0=K+2, VGPR1=K+3.  -> float2 load at
// X[row*K + kb] with kb = k + (lane>=16 ? 2 : 0).
// B (4x16 f32) mirrors A with N in place of M.  C/D: VGPR r -> M=r+half*8,
// N = lane&15.
// ---------------------------------------------------------------------------
__global__ void __launch_bounds__(256)
gemm_bias_wmma(const float* __restrict__ X, const float* __restrict__ W,
               const float* __restrict__ bias, float* __restrict__ Y,
               int K, int n_rows)
{
    const int wave    = (blockIdx.x * blockDim.x + threadIdx.x) >> 5;
    const int lane    = threadIdx.x & 31;
    const int n_tiles = (n_rows + 15) >> 4;
    if (wave >= n_tiles) return;               // wave-uniform exit: EXEC stays all-1s

    const int half = lane >> 4;                // 0: lanes 0-15, 1: lanes 16-31
    const int mrow = lane & 15;
    const int arow = min(wave * 16 + mrow, n_rows - 1);
    const float* __restrict__ xrow = X + (size_t)arow * (size_t)K;

    v8f c0 = {};
    v8f c1 = {};

    for (int k = 0; k < K; k += 4) {
        const int kb = k + half * 2;
        v2f a = *(const v2f*)(xrow + kb);

        v2f b0, b1;                            // W is row-major [K,32], tiny -> L0/L2 resident
        b0.x = W[(size_t)kb * 32 + mrow];
        b0.y = W[(size_t)(kb + 1) * 32 + mrow];
        b1.x = W[(size_t)kb * 32 + 16 + mrow];
        b1.y = W[(size_t)(kb + 1) * 32 + 16 + mrow];

        c0 = __builtin_amdgcn_wmma_f32_16x16x4_f32(false, a, false, b0,
                                                   (short)0, c0, false, false);
        c1 = __builtin_amdgcn_wmma_f32_16x16x4_f32(false, a, false, b1,
                                                   (short)0, c1, false, false);
    }

    const float bv0 = bias[mrow];
    const float bv1 = bias[16 + mrow];
#pragma unroll
    for (int r = 0; r < 8; ++r) {
        const int orow = wave * 16 + r + half * 8;
        if (orow < n_rows) {
            Y[(size_t)orow * 32 + mrow]      = c0[r] + bv0;
            Y[(size_t)orow * 32 + 16 + mrow] = c1[r] + bv1;
        }
    }
}

// Zero accumulators + degree counter (re-run every call: graph replays).
__global__ void init_zero(float* __restrict__ cnt, float* __restrict__ accB,
                          float* __restrict__ out, int n_nodes)
{
    const int i     = blockIdx.x * blockDim.x + threadIdx.x;
    const int total = n_nodes * 32;
    if (i < total) { accB[i] = 0.0f; out[i] = 0.0f; }
    if (i < n_nodes) cnt[i] = 0.0f;
}

__global__ void degree_kernel(const int* __restrict__ dst,
                              float* __restrict__ cnt, int n_edges)
{
    const int e = blockIdx.x * blockDim.x + threadIdx.x;
    if (e < n_edges) atomicAdd(&cnt[dst[e]], 1.0f);
}

// One (edge, feature) per thread; lane = feature so the wave's 32 atomics hit
// one 128B line of the destination node (L2-resident: 12.8MB << 192MB L2).
__global__ void scatter_accum(const int* __restrict__ src, const int* __restrict__ dst,
                              const float* __restrict__ H, float* __restrict__ Acc,
                              int n_edges)
{
    const int t = blockIdx.x * blockDim.x + threadIdx.x;
    const int e = t >> 5;
    const int f = t & 31;
    if (e < n_edges) {
        const int s = src[e];
        const int d = dst[e];
        atomicAdd(&Acc[(size_t)d * 32 + f], H[(size_t)s * 32 + f]);
    }
}

// A[i] = relu(B[i] / max(cnt,1))   (mean with PyG isolated-node -> 0 behavior)
__global__ void finalize_relu(const float* __restrict__ B, const float* __restrict__ cnt,
                              float* __restrict__ A, int n_nodes)
{
    const int i = blockIdx.x * blockDim.x + threadIdx.x;
    if (i < n_nodes * 32) {
        const float c = cnt[i >> 5];
        const float v = B[i] / fmaxf(c, 1.0f);
        A[i] = fmaxf(v, 0.0f);
    }
}

// out[i] /= max(cnt,1)  in place
__global__ void finalize_div(float* __restrict__ out, const float* __restrict__ cnt,
                             int n_nodes)
{
    const int i = blockIdx.x * blockDim.x + threadIdx.x;
    if (i < n_nodes * 32) {
        out[i] = out[i] / fmaxf(cnt[i >> 5], 1.0f);
    }
}

extern "C" void kernel_launch(void* const* d_in, const int* in_sizes, int n_in,
                              void* d_out, int out_size, void* d_ws, size_t ws_size,
                              hipStream_t stream)
{
    const float* x    = (const float*)d_in[0];
    const int*   eidx = (const int*)d_in[1];
    const float* W1   = (const float*)d_in[2];
    const float* b1   = (const float*)d_in[3];
    const float* W2   = (const float*)d_in[4];
    const float* b2   = (const float*)d_in[5];
    float*       out  = (float*)d_out;

    const int n_nodes = in_sizes[0] / 128;
    const int n_edges = in_sizes[1] / 2;
    const int* src = eidx;             // edge_index[0]
    const int* dst = eidx + n_edges;   // edge_index[1]

    // Workspace layout: A (h buffer), B (scatter accumulator / h2), cnt.
    float* A   = (float*)d_ws;
    float* B   = A + (size_t)n_nodes * 32;
    float* cnt = B + (size_t)n_nodes * 32;

    const int TPB = 256;
    const int n_feat_elems   = n_nodes * 32;
    const int grid_feat      = (n_feat_elems + TPB - 1) / TPB;
    const int grid_edges     = (n_edges + TPB - 1) / TPB;
    const int grid_edge_feat = (int)(((size_t)n_edges * 32 + TPB - 1) / TPB);
    const int n_tiles        = (n_nodes + 15) / 16;
    const int grid_gemm      = (n_tiles + 7) / 8;   // 8 waves per 256-thread block

    // 1. zero cnt, B, out
    init_zero<<<grid_feat, TPB, 0, stream>>>(cnt, B, out, n_nodes);
    // 2. degrees
    degree_kernel<<<grid_edges, TPB, 0, stream>>>(dst, cnt, n_edges);
    // 3. A = x @ W1 + b1
    gemm_bias_wmma<<<grid_gemm, TPB, 0, stream>>>(x, W1, b1, A, 128, n_nodes);
    // 4. B += A[src] at dst
    scatter_accum<<<grid_edge_feat, TPB, 0, stream>>>(src, dst, A, B, n_edges);
    // 5. A = relu(B / max(cnt,1))
    finalize_relu<<<grid_feat, TPB, 0, stream>>>(B, cnt, A, n_nodes);
    // 6. B = A @ W2 + b2
    gemm_bias_wmma<<<grid_gemm, TPB, 0, stream>>>(A, W2, b2, B, 32, n_nodes);
    // 7. out += B[src] at dst
    scatter_accum<<<grid_edge_feat, TPB, 0, stream>>>(src, dst, B, out, n_edges);
    // 8. out /= max(cnt,1)
    finalize_div<<<grid_feat, TPB, 0, stream>>>(out, cnt, n_nodes);
}